// QuLinearAngleEmbedding_17454747091659
// MI455X (gfx1250) — compile-verified
//
#include <hip/hip_runtime.h>
#include <hip/hip_bf16.h>

// 11-qubit batched state-vector simulator + partial-trace via f32 WMMA.
// One workgroup (256 threads = 8 wave32) per batch element; state lives in LDS.

typedef float v2f __attribute__((ext_vector_type(2)));
typedef float v8f __attribute__((ext_vector_type(8)));

#define NQ     11
#define QDIM   2048       // 1 << NQ
#define NLAYER 5

__global__ __launch_bounds__(256)
void qsim_wmma_kernel(const float* __restrict__ x,   // [B, 1, 11]
                      const float* __restrict__ w,   // [33 * 6]
                      float* __restrict__ out)       // [B, 32, 32] complex64 -> float pairs
{
    __shared__ float sRe[QDIM];
    __shared__ float sIm[QDIM];
    __shared__ float sU[NQ][8];      // fused 2x2 complex gate per qubit per layer
    __shared__ float sC[NQ], sS[NQ]; // embedding cos/sin

    const int tid = threadIdx.x;
    const int b   = blockIdx.x;

    // ---- angle embedding as product state: amp(idx) = prod_q (bit_q ? sin : cos)(x_q/2)
    if (tid < NQ) {
        float h = 0.5f * x[b * NQ + tid];
        sC[tid] = cosf(h);
        sS[tid] = sinf(h);
    }
    __syncthreads();

    #pragma unroll
    for (int t = 0; t < 8; ++t) {
        int idx = tid + t * 256;
        float p = 1.0f;
        #pragma unroll
        for (int q = 0; q < NQ; ++q) {
            int bit = (idx >> (10 - q)) & 1;
            p *= bit ? sS[q] : sC[q];
        }
        sRe[idx] = p;
        sIm[idx] = 0.0f;
    }
    __syncthreads();

    // ---- 6 fused YZY layers; CNOT ring after layers 0..4
    for (int layer = 0; layer <= NLAYER; ++layer) {
        // fused U = Ry(p2) * Rz(p1) * Ry(p0), one per qubit, computed by threads 0..10
        if (tid < NQ) {
            const float* wp = w + layer * 3 * NQ + tid * 3;
            float c0 = cosf(0.5f * wp[0]), s0 = sinf(0.5f * wp[0]);
            float cb = cosf(0.5f * wp[1]), sb = sinf(0.5f * wp[1]);
            float c2 = cosf(0.5f * wp[2]), s2 = sinf(0.5f * wp[2]);
            float a = c2 * c0, d = s2 * s0, e = s2 * c0, f = c2 * s0;
            // em = cb - i sb, ep = conj(em)
            sU[tid][0] = (a - d) * cb;   sU[tid][1] = -(a + d) * sb;  // U00
            sU[tid][2] = -(f + e) * cb;  sU[tid][3] = (f - e) * sb;   // U01
            sU[tid][4] = (e + f) * cb;   sU[tid][5] = (f - e) * sb;   // U10
            sU[tid][6] = (a - d) * cb;   sU[tid][7] = (a + d) * sb;   // U11
        }
        __syncthreads();

        for (int q = 0; q < NQ; ++q) {
            const int mask = 1 << (10 - q);
            const float u00r = sU[q][0], u00i = sU[q][1];
            const float u01r = sU[q][2], u01i = sU[q][3];
            const float u10r = sU[q][4], u10i = sU[q][5];
            const float u11r = sU[q][6], u11i = sU[q][7];
            #pragma unroll
            for (int t = 0; t < 4; ++t) {
                int p  = tid + t * 256;          // 1024 amp pairs
                int lo = p & (mask - 1);
                int i0 = ((p ^ lo) << 1) | lo;
                int i1 = i0 | mask;
                float a0r = sRe[i0], a0i = sIm[i0];
                float a1r = sRe[i1], a1i = sIm[i1];
                sRe[i0] = u00r * a0r - u00i * a0i + u01r * a1r - u01i * a1i;
                sIm[i0] = u00r * a0i + u00i * a0r + u01r * a1i + u01i * a1r;
                sRe[i1] = u10r * a0r - u10i * a0i + u11r * a1r - u11i * a1i;
                sIm[i1] = u10r * a0i + u10i * a0r + u11r * a1i + u11i * a1r;
            }
            __syncthreads();
        }

        if (layer < NLAYER) {
            // ring of CNOTs == one GF(2)-linear basis permutation, applied as a gather:
            // psi_new[x] = psi_old[g(x)], qubit q at bit (10-q).
            float rr[8], ri[8];
            #pragma unroll
            for (int t = 0; t < 8; ++t) {
                unsigned xI  = (unsigned)(tid + t * 256);
                unsigned gx  = (xI ^ (xI >> 1)) & 0x1FFu;
                unsigned b0  = xI & 1u;            // x10
                unsigned b9  = (xI >> 9) & 1u;     // x1
                unsigned b10 = (xI >> 10) & 1u;    // x0
                gx |= ((b9 ^ b10 ^ b0) << 9) | ((b10 ^ b0) << 10);
                rr[t] = sRe[gx];
                ri[t] = sIm[gx];
            }
            __syncthreads();
            #pragma unroll
            for (int t = 0; t < 8; ++t) {
                int idx = tid + t * 256;
                sRe[idx] = rr[t];
                sIm[idx] = ri[t];
            }
            __syncthreads();
        }
    }

    // ---- partial trace: out = A^T * conj(A), A = psi.reshape(64, 32)
    // Re = Ar^T Ar + Ai^T Ai ; Im = Ai^T Ar - Ar^T Ai.
    // 8 wave32s: wave = part + 2*JT + 4*KT; each wave owns one 16x16 tile x {Re,Im}.
    const int lane = tid & 31;
    const int wv   = tid >> 5;
    const int part = wv & 1;
    const int JT   = (wv >> 1) & 1;
    const int KT   = (wv >> 2) & 1;
    const int m    = lane & 15;
    const int khi  = (lane >> 4) << 1;      // lanes 16-31 carry K pairs 2/3
    const int aCol = JT * 16 + m;           // A-frag: Ahat[m][k] = P[k*32 + aCol]
    const int bCol = KT * 16 + m;           // B-frag: B[k][n]    = Q[k*32 + bCol]

    v8f accRe = {};
    v8f accP  = {};
    v8f accM  = {};

    if (part == 0) {
        for (int kb = 0; kb < 64; kb += 4) {
            int k0 = (kb + khi) * 32, k1 = k0 + 32;
            v2f arA, aiA, arB, aiB;
            arA[0] = sRe[k0 + aCol]; arA[1] = sRe[k1 + aCol];
            aiA[0] = sIm[k0 + aCol]; aiA[1] = sIm[k1 + aCol];
            arB[0] = sRe[k0 + bCol]; arB[1] = sRe[k1 + bCol];
            aiB[0] = sIm[k0 + bCol]; aiB[1] = sIm[k1 + bCol];
            accRe = __builtin_amdgcn_wmma_f32_16x16x4_f32(false, arA, false, arB,
                                                          (short)0, accRe, false, false);
            accRe = __builtin_amdgcn_wmma_f32_16x16x4_f32(false, aiA, false, aiB,
                                                          (short)0, accRe, false, false);
        }
    } else {
        for (int kb = 0; kb < 64; kb += 4) {
            int k0 = (kb + khi) * 32, k1 = k0 + 32;
            v2f arA, aiA, arB, aiB;
            arA[0] = sRe[k0 + aCol]; arA[1] = sRe[k1 + aCol];
            aiA[0] = sIm[k0 + aCol]; aiA[1] = sIm[k1 + aCol];
            arB[0] = sRe[k0 + bCol]; arB[1] = sRe[k1 + bCol];
            aiB[0] = sIm[k0 + bCol]; aiB[1] = sIm[k1 + bCol];
            accP = __builtin_amdgcn_wmma_f32_16x16x4_f32(false, aiA, false, arB,
                                                         (short)0, accP, false, false);
            accM = __builtin_amdgcn_wmma_f32_16x16x4_f32(false, arA, false, aiB,
                                                         (short)0, accM, false, false);
        }
    }

    // C/D layout: VGPR r -> row M=r (lanes 0-15) or M=r+8 (lanes 16-31), col = lane%16.
    float* ob = out + (size_t)b * (32 * 32 * 2);
    const int colk = KT * 16 + (lane & 15);
    const int rofs = JT * 16 + ((lane >> 4) << 3);
    if (part == 0) {
        #pragma unroll
        for (int r = 0; r < 8; ++r)
            ob[((rofs + r) * 32 + colk) * 2 + 0] = accRe[r];
    } else {
        #pragma unroll
        for (int r = 0; r < 8; ++r)
            ob[((rofs + r) * 32 + colk) * 2 + 1] = accP[r] - accM[r];
    }
}

extern "C" void kernel_launch(void* const* d_in, const int* in_sizes, int n_in,
                              void* d_out, int out_size, void* d_ws, size_t ws_size,
                              hipStream_t stream) {
    const float* x = (const float*)d_in[0];   // [B, 1, 11] float32
    const float* w = (const float*)d_in[1];   // [198] float32
    // d_in[2] = dimA (int, ==5) : layout constants baked in (dA=32, dB=64)
    float* out = (float*)d_out;               // [B, 32, 32] complex64 as float pairs

    int B = in_sizes[0] / NQ;                 // 16
    qsim_wmma_kernel<<<B, 256, 0, stream>>>(x, w, out);
}